// RetraceLoss_63900523430190
// MI455X (gfx1250) — compile-verified
//
#include <hip/hip_runtime.h>
#include <hip/hip_bf16.h>

// Retrace loss on MI455X (gfx1250).
//
// W[i,j,b] = gamma^(j-i) exp(P[j,b]-P[i,b]) (j>=i) factorizes through
// u[t,b] = gamma^t exp(P[t,b]):
//   retrace[i,b] = ( SA[i,b] + tqc[i,b]*SU[i,b] ) / u[i,b]
//   SA = Upper @ (u*A),  SU = Upper @ u,  Upper[i,j] = (j>=i)
// Upper is batch-independent -> suffix sums are a 256x256 @ 256x1024 f32 GEMM
// pair on V_WMMA_F32_16X16X4_F32 (A synthesized in-register; B staged in LDS).
// Collapses the reference's O(T^2 B) einsum + 268MB W tensor to 6MB of reads
// + ~134M f32 MACs (triangular K-skip removes the zero half).

#define TT 256
#define BB 1024
#define LOG_GAMMA (-0.05129329438755058f)   // log(0.95)

typedef float v2f __attribute__((ext_vector_type(2)));
typedef float v8f __attribute__((ext_vector_type(8)));

// ---------------------------------------------------------------- kernel A --
// One thread per column b: exclusive prefix scan, emit u and u*A.
__global__ void retrace_scan_kernel(const float* __restrict__ tgt_log,
                                    const float* __restrict__ org_log,
                                    const float* __restrict__ rewards,
                                    const float* __restrict__ tq_traj,
                                    float* __restrict__ g_u,
                                    float* __restrict__ g_uA) {
    const int b = blockIdx.x * blockDim.x + threadIdx.x;   // 0..1023
    float s = 0.0f;                                        // P[t] + t*log(gamma)
    #pragma unroll 4
    for (int t = 0; t < TT; ++t) {
        const int idx = t * BB + b;
        const float lw = fmaxf(tgt_log[idx] - org_log[idx], 0.0f);
        const float ut = __expf(s);
        g_u[idx]  = ut;
        g_uA[idx] = ut * (rewards[idx] - tq_traj[idx]);
        s += lw + LOG_GAMMA;
    }
}

// ---------------------------------------------------------------- kernel B --
// 64 blocks (16 columns each) x 512 threads (16 waves).
// Wave I owns output rows [16I, 16I+16): triangular skip -> K starts at 16I.
__global__ void __launch_bounds__(512)
retrace_wmma_kernel(const float* __restrict__ g_u,
                    const float* __restrict__ g_uA,
                    const float* __restrict__ tq_curr,
                    const float* __restrict__ q_traj,
                    float* __restrict__ partials) {
    __shared__ float u_lds[TT * 16];    // 16 KB: u[k, n]   for this column tile
    __shared__ float uA_lds[TT * 16];   // 16 KB: u*A[k, n]
    __shared__ float red[512];

    const int n0 = blockIdx.x * 16;     // first column of this tile

    // Stage the whole 256x16 column chunk once (coalesced 64B rows).
    for (int i = threadIdx.x; i < TT * 16; i += 512) {
        const int k = i >> 4, n = i & 15;
        u_lds[i]  = g_u[k * BB + n0 + n];
        uA_lds[i] = g_uA[k * BB + n0 + n];
    }
    __syncthreads();

    const int lane  = threadIdx.x & 31;
    // Wave index as an SGPR: scalar loop bounds, no EXEC-masked loop around
    // the WMMAs (EXEC must be all-ones for WMMA).
    const int waveI = __builtin_amdgcn_readfirstlane(threadIdx.x >> 5); // 0..15
    const int i0    = waveI << 4;
    const int m     = lane & 15;            // A: row / B,C,D: column
    const int khalf = (lane >> 4) << 1;     // 0 or 2 (K split across half-waves)

    v8f accA = {};                          // SA tile accumulator
    v8f accU = {};                          // SU tile accumulator

    // ---- Prologue: the 4 K-steps on the diagonal block carry the mask. ----
    #pragma unroll
    for (int kk = 0; kk < 16; kk += 4) {
        v2f a;
        a.x = ((kk + khalf)     >= m) ? 1.0f : 0.0f;
        a.y = ((kk + khalf + 1) >= m) ? 1.0f : 0.0f;
        const int krow = (i0 + kk + khalf) << 4;
        v2f bU, bA;
        bU.x = u_lds[krow + m];   bU.y = u_lds[krow + 16 + m];
        bA.x = uA_lds[krow + m];  bA.y = uA_lds[krow + 16 + m];
        accA = __builtin_amdgcn_wmma_f32_16x16x4_f32(false, a, false, bA,
                                                     (short)0, accA, false, false);
        accU = __builtin_amdgcn_wmma_f32_16x16x4_f32(false, a, false, bU,
                                                     (short)0, accU, false, false);
    }

    // ---- Steady state: A-tile is all ones. (256-(i0+16))/8 is integral for
    //      every wave, so the x2 unroll is exact. ----
    v2f ones; ones.x = 1.0f; ones.y = 1.0f;
    for (int k0 = i0 + 16; k0 < TT; k0 += 8) {
        const int krow0 = (k0 + khalf) << 4;
        const int krow1 = krow0 + 64;       // +4 K-rows
        v2f bU0, bA0, bU1, bA1;
        bU0.x = u_lds[krow0 + m];   bU0.y = u_lds[krow0 + 16 + m];
        bA0.x = uA_lds[krow0 + m];  bA0.y = uA_lds[krow0 + 16 + m];
        bU1.x = u_lds[krow1 + m];   bU1.y = u_lds[krow1 + 16 + m];
        bA1.x = uA_lds[krow1 + m];  bA1.y = uA_lds[krow1 + 16 + m];
        accA = __builtin_amdgcn_wmma_f32_16x16x4_f32(false, ones, false, bA0,
                                                     (short)0, accA, false, false);
        accU = __builtin_amdgcn_wmma_f32_16x16x4_f32(false, ones, false, bU0,
                                                     (short)0, accU, false, false);
        accA = __builtin_amdgcn_wmma_f32_16x16x4_f32(false, ones, false, bA1,
                                                     (short)0, accA, false, false);
        accU = __builtin_amdgcn_wmma_f32_16x16x4_f32(false, ones, false, bU1,
                                                     (short)0, accU, false, false);
    }

    // Epilogue: C/D layout — VGPR r: lanes 0-15 -> M=r, lanes 16-31 -> M=r+8.
    float lsum = 0.0f;
    const int gcol = n0 + m;
    const int mhi  = (lane >> 4) << 3;      // +8 for upper half-wave
    #pragma unroll
    for (int r = 0; r < 8; ++r) {
        const int gi   = i0 + r + mhi;
        const float sa = accA[r];
        const float su = accU[r];
        const float uinv = __builtin_amdgcn_rcpf(u_lds[(gi << 4) + m]);
        const float retrace = (sa + tq_curr[gi * BB + gcol] * su) * uinv;
        const float d = q_traj[gi * BB + gcol] - retrace;
        lsum += d * d;
    }

    // Deterministic fixed-order block reduction.
    red[threadIdx.x] = lsum;
    __syncthreads();
    for (int s = 256; s > 0; s >>= 1) {
        if (threadIdx.x < s) red[threadIdx.x] += red[threadIdx.x + s];
        __syncthreads();
    }
    if (threadIdx.x == 0) partials[blockIdx.x] = red[0];
}

// ---------------------------------------------------------------- kernel C --
__global__ void retrace_finalize_kernel(const float* __restrict__ partials,
                                        float* __restrict__ out) {
    if (threadIdx.x == 0 && blockIdx.x == 0) {
        float s = 0.0f;
        for (int i = 0; i < 64; ++i) s += partials[i];   // fixed order
        out[0] = s * (1.0f / (float)(TT * BB));
    }
}

// ------------------------------------------------------------------- host --
extern "C" void kernel_launch(void* const* d_in, const int* in_sizes, int n_in,
                              void* d_out, int out_size, void* d_ws, size_t ws_size,
                              hipStream_t stream) {
    (void)in_sizes; (void)n_in; (void)out_size; (void)ws_size;
    // setup_inputs() order:
    const float* q_traj  = (const float*)d_in[0];  // state_trajectory_action_values
    const float* tq_traj = (const float*)d_in[1];  // target_state_trajectory_action_values
    const float* tq_curr = (const float*)d_in[2];  // target_state_current_action_values
    const float* rewards = (const float*)d_in[3];
    const float* org_log = (const float*)d_in[4];
    const float* tgt_log = (const float*)d_in[5];

    float* w        = (float*)d_ws;
    float* g_u      = w;                 // T*B floats
    float* g_uA     = w + TT * BB;       // T*B floats
    float* partials = w + 2 * TT * BB;   // 64 floats

    retrace_scan_kernel<<<dim3(BB / 256), dim3(256), 0, stream>>>(
        tgt_log, org_log, rewards, tq_traj, g_u, g_uA);

    retrace_wmma_kernel<<<dim3(BB / 16), dim3(512), 0, stream>>>(
        g_u, g_uA, tq_curr, q_traj, partials);

    retrace_finalize_kernel<<<dim3(1), dim3(64), 0, stream>>>(
        partials, (float*)d_out);
}